// PersonaGNN_80796924772538
// MI455X (gfx1250) — compile-verified
//
#include <hip/hip_runtime.h>
#include <math.h>

typedef __attribute__((ext_vector_type(2))) float v2f;
typedef __attribute__((ext_vector_type(8))) float v8f;

#define DFEAT 128
#define NEG_SLOPE 0.2f

// ---------------------------------------------------------------------------
// GEMM: h[N,128] = in[N,128] @ W[128,128] using V_WMMA_F32_16X16X4_F32.
// One wave (32 lanes) owns a 16-row tile and all 8 column tiles (16 cols each).
// W is staged in LDS (64KB of the 320KB WGP pool).
// A 16x4 layout (ISA 7.12.2): lanes 0-15 hold K=k,k+1; lanes 16-31 hold K=k+2,k+3.
// C/D layout: VGPR r -> M=r (lanes 0-15) / M=r+8 (lanes 16-31), N=lane&15.
// ---------------------------------------------------------------------------
__global__ __launch_bounds__(256) void gemm_wmma_f32(
    const float* __restrict__ in, const float* __restrict__ W,
    float* __restrict__ h, int n_rows)
{
    __shared__ float Wlds[DFEAT * DFEAT];
    for (int i = threadIdx.x; i < DFEAT * DFEAT; i += 256) Wlds[i] = W[i];
    __syncthreads();

    const int wave  = threadIdx.x >> 5;
    const int lane  = threadIdx.x & 31;
    const int mtile = blockIdx.x * 8 + wave;
    if (mtile * 16 >= n_rows) return;          // wave-uniform: EXEC stays all-1s for WMMA
    const int m0  = mtile * 16;
    const int hi  = lane >> 4;                 // 0: lanes 0-15, 1: lanes 16-31
    const int l15 = lane & 15;

    v8f acc[8] = {};
    const float* arow = in + (size_t)(m0 + l15) * DFEAT;

    for (int k = 0; k < DFEAT; k += 4) {
        v2f a;
        a.x = arow[k + 2 * hi];
        a.y = arow[k + 2 * hi + 1];
#pragma unroll
        for (int t = 0; t < 8; ++t) {
            const int col = t * 16 + l15;
            v2f b;
            b.x = Wlds[(k + 2 * hi) * DFEAT + col];
            b.y = Wlds[(k + 2 * hi + 1) * DFEAT + col];
            acc[t] = __builtin_amdgcn_wmma_f32_16x16x4_f32(
                false, a, false, b, (short)0, acc[t], false, false);
        }
    }

#pragma unroll
    for (int t = 0; t < 8; ++t) {
        const int col = t * 16 + l15;
#pragma unroll
        for (int r = 0; r < 8; ++r) {
            h[(size_t)(m0 + r + 8 * hi) * DFEAT + col] = acc[t][r];
        }
    }
}

// ---------------------------------------------------------------------------
// Per-node attention logits: s[i] = dot(h[i], a_src), d[i] = dot(h[i], a_dst).
// One wave per node, float4 per lane, shuffle reduction.
// ---------------------------------------------------------------------------
__global__ __launch_bounds__(256) void node_logits(
    const float* __restrict__ h, const float* __restrict__ a_src,
    const float* __restrict__ a_dst, float* __restrict__ sbuf,
    float* __restrict__ dbuf, int n)
{
    const int wid  = (int)((blockIdx.x * 256 + threadIdx.x) >> 5);
    const int lane = threadIdx.x & 31;
    if (wid >= n) return;
    const float4 hv = ((const float4*)(h + (size_t)wid * DFEAT))[lane];
    const float4 as = ((const float4*)a_src)[lane];
    const float4 ad = ((const float4*)a_dst)[lane];
    float ss = hv.x * as.x + hv.y * as.y + hv.z * as.z + hv.w * as.w;
    float dd = hv.x * ad.x + hv.y * ad.y + hv.z * ad.z + hv.w * ad.w;
#pragma unroll
    for (int off = 16; off > 0; off >>= 1) {
        ss += __shfl_xor(ss, off, 32);
        dd += __shfl_xor(dd, off, 32);
    }
    if (lane == 0) { sbuf[wid] = ss; dbuf[wid] = dd; }
}

__device__ __forceinline__ void atomicMaxF(float* addr, float val) {
    unsigned int* ua = (unsigned int*)addr;
    unsigned int old = *ua;
    while (true) {
        float oldf = __uint_as_float(old);
        if (oldf >= val) break;
        unsigned int assumed = old;
        old = atomicCAS(ua, assumed, __float_as_uint(val));
        if (old == assumed) break;
    }
}

// Pass 1: e = leaky_relu(s[src] + d[dst]); segment max over dst.
__global__ __launch_bounds__(256) void edge_logit_max(
    const int* __restrict__ ei, int E, int n,
    const float* __restrict__ sbuf, const float* __restrict__ dbuf,
    float* __restrict__ ebuf, float* __restrict__ mmax)
{
    const int i   = blockIdx.x * 256 + threadIdx.x;
    const int tot = E + n;
    if (i >= tot) return;
    int sIdx, dIdx;
    if (i < E) { sIdx = ei[i]; dIdx = ei[E + i]; }
    else       { sIdx = dIdx = i - E; }           // self-loop
    float ev = sbuf[sIdx] + dbuf[dIdx];
    ev = ev > 0.0f ? ev : NEG_SLOPE * ev;
    ebuf[i] = ev;
    atomicMaxF(&mmax[dIdx], ev);
}

// Pass 2: p = exp(e - m[dst]); segment sum over dst. Stores p back into ebuf.
__global__ __launch_bounds__(256) void edge_expsum(
    const int* __restrict__ ei, int E, int n,
    float* __restrict__ ebuf, const float* __restrict__ mmax,
    float* __restrict__ denom)
{
    const int i   = blockIdx.x * 256 + threadIdx.x;
    const int tot = E + n;
    if (i >= tot) return;
    const int dIdx = (i < E) ? ei[E + i] : (i - E);
    const float pv = expf(ebuf[i] - mmax[dIdx]);
    ebuf[i] = pv;
    atomicAdd(&denom[dIdx], pv);
}

// Pass 3: acc[dst] += (p / denom[dst]) * h[src]. One wave per edge, float4 lanes.
__global__ __launch_bounds__(256) void edge_scatter(
    const int* __restrict__ ei, int E, int n,
    const float* __restrict__ ebuf, const float* __restrict__ denom,
    const float* __restrict__ h, float* __restrict__ acc)
{
    const int wid  = (int)((blockIdx.x * 256 + threadIdx.x) >> 5);
    const int lane = threadIdx.x & 31;
    const int tot  = E + n;
    if (wid >= tot) return;
    int sIdx, dIdx;
    if (wid < E) { sIdx = ei[wid]; dIdx = ei[E + wid]; }
    else         { sIdx = dIdx = wid - E; }
    const float alpha = ebuf[wid] / denom[dIdx];
    const float4 hv = ((const float4*)(h + (size_t)sIdx * DFEAT))[lane];
    float* o = acc + (size_t)dIdx * DFEAT + lane * 4;
    atomicAdd(o + 0, alpha * hv.x);
    atomicAdd(o + 1, alpha * hv.y);
    atomicAdd(o + 2, alpha * hv.z);
    atomicAdd(o + 3, alpha * hv.w);
}

// Per-layer init: zero accumulator, -inf segment maxima, zero denominators/colsum.
__global__ __launch_bounds__(256) void init_layer(
    float* __restrict__ acc, float* __restrict__ mmax,
    float* __restrict__ denom, float* __restrict__ colsum, int n)
{
    const int i = blockIdx.x * 256 + threadIdx.x;
    if (i < n * DFEAT) acc[i] = 0.0f;
    if (i < n) { mmax[i] = -__builtin_inff(); denom[i] = 0.0f; }
    if (i < DFEAT) colsum[i] = 0.0f;
}

__global__ __launch_bounds__(256) void bias_act(
    float* __restrict__ acc, const float* __restrict__ b, int total, int do_relu)
{
    const int i = blockIdx.x * 256 + threadIdx.x;
    if (i >= total) return;
    float v = acc[i] + b[i & (DFEAT - 1)];
    if (do_relu) v = fmaxf(v, 0.0f);
    acc[i] = v;
}

// Column reduction: colsum[j] += sum_i acc[i][j] (strided partials + atomics).
__global__ __launch_bounds__(256) void col_reduce(
    const float* __restrict__ acc, float* __restrict__ colsum, int n)
{
    const int j       = threadIdx.x & (DFEAT - 1);
    const int gtid    = blockIdx.x * 256 + threadIdx.x;
    const int slice   = gtid / DFEAT;
    const int nslices = (gridDim.x * 256) / DFEAT;
    float s = 0.0f;
    for (int r = slice; r < n; r += nslices) s += acc[(size_t)r * DFEAT + j];
    atomicAdd(&colsum[j], s);
}

__global__ void finalize_mean(
    const float* __restrict__ colsum, const float* __restrict__ b,
    float* __restrict__ out, int n)
{
    const int j = threadIdx.x;
    if (j < DFEAT) out[j] = colsum[j] / (float)n + b[j];
}

// ---------------------------------------------------------------------------
extern "C" void kernel_launch(void* const* d_in, const int* in_sizes, int n_in,
                              void* d_out, int out_size, void* d_ws, size_t ws_size,
                              hipStream_t stream) {
    const float* x   = (const float*)d_in[0];
    const int*   ei  = (const int*)d_in[1];
    const float* W1  = (const float*)d_in[2];
    const float* as1 = (const float*)d_in[3];
    const float* ad1 = (const float*)d_in[4];
    const float* b1  = (const float*)d_in[5];
    const float* W2  = (const float*)d_in[6];
    const float* as2 = (const float*)d_in[7];
    const float* ad2 = (const float*)d_in[8];
    const float* b2  = (const float*)d_in[9];
    float* out = (float*)d_out;

    const int n   = in_sizes[0] / DFEAT;   // 50000
    const int E   = in_sizes[1] / 2;       // 800000
    const int tot = E + n;                 // edges incl. self-loops

    // Workspace carve-up (floats)
    float* ws = (float*)d_ws;
    size_t off = 0;
    float* h      = ws + off; off += (size_t)n * DFEAT;
    float* acc    = ws + off; off += (size_t)n * DFEAT;
    float* sbuf   = ws + off; off += n;
    float* dbuf   = ws + off; off += n;
    float* mmax   = ws + off; off += n;
    float* denom  = ws + off; off += n;
    float* ebuf   = ws + off; off += tot;
    float* colsum = ws + off; off += DFEAT;

    const dim3 blk(256);
    const int gInit = (n * DFEAT + 255) / 256;
    const int gGemm = ((n + 15) / 16 + 7) / 8;       // 16-row tiles, 8 waves/block
    const int gNode = ((n * 32) + 255) / 256;        // one wave per node
    const int gEdge = (tot + 255) / 256;
    const int gScat = ((tot * 32) + 255) / 256;      // one wave per edge

    // -------- Layer 1 --------
    init_layer<<<gInit, blk, 0, stream>>>(acc, mmax, denom, colsum, n);
    gemm_wmma_f32<<<gGemm, blk, 0, stream>>>(x, W1, h, n);
    node_logits<<<gNode, blk, 0, stream>>>(h, as1, ad1, sbuf, dbuf, n);
    edge_logit_max<<<gEdge, blk, 0, stream>>>(ei, E, n, sbuf, dbuf, ebuf, mmax);
    edge_expsum<<<gEdge, blk, 0, stream>>>(ei, E, n, ebuf, mmax, denom);
    edge_scatter<<<gScat, blk, 0, stream>>>(ei, E, n, ebuf, denom, h, acc);
    bias_act<<<gInit, blk, 0, stream>>>(acc, b1, n * DFEAT, 1);   // relu(acc + b1)

    // -------- Layer 2 (input = acc; gemm must read acc before re-init) --------
    gemm_wmma_f32<<<gGemm, blk, 0, stream>>>(acc, W2, h, n);
    init_layer<<<gInit, blk, 0, stream>>>(acc, mmax, denom, colsum, n);
    node_logits<<<gNode, blk, 0, stream>>>(h, as2, ad2, sbuf, dbuf, n);
    edge_logit_max<<<gEdge, blk, 0, stream>>>(ei, E, n, sbuf, dbuf, ebuf, mmax);
    edge_expsum<<<gEdge, blk, 0, stream>>>(ei, E, n, ebuf, mmax, denom);
    edge_scatter<<<gScat, blk, 0, stream>>>(ei, E, n, ebuf, denom, h, acc);

    // mean over nodes + bias
    col_reduce<<<256, blk, 0, stream>>>(acc, colsum, n);
    finalize_mean<<<1, 128, 0, stream>>>(colsum, b2, out, n);
}